// LSDS_2559800509159
// MI455X (gfx1250) — compile-verified
//
#include <hip/hip_runtime.h>

// ---------------- problem constants ----------------
#define Bq   4
#define Cq   16
#define Hq   512
#define Wq   512
#define HW   (Hq*Wq)          // 262144
#define ROWS (Bq*Cq)          // 64
#define HID  64
#define OUTC 16
#define KSEL (HW/4)           // 65536
#define OH   256
#define OW   256
#define EPSF 1e-5f

// ---------------- workspace layout (float offsets) ----------------
#define OFF_ACC1   ((size_t)0)            // 64*64
#define OFF_H      ((size_t)4096)         // 64*64
#define OFF_RSUM   ((size_t)8192)         // 64
#define OFF_RSQ    ((size_t)8256)         // 64
#define OFF_HIST   ((size_t)8320)         // 4 passes * 4 batches * 256
#define OFF_STATE  ((size_t)12416)        // 4 batches * 4 u32 {prefix,kRem,gacc,pad}
#define OFF_CGT    ((size_t)12432)        // 4*256
#define OFF_CEQ    ((size_t)13456)        // 4*256
#define OFF_PGT    ((size_t)14480)        // 4*256
#define OFF_PEQ    ((size_t)15504)        // 4*256
#define OFF_BNS    ((size_t)16528)        // 16 sums + 16 sumsq
#define OFF_M2     ((size_t)16640)        // 64*HW
#define OFF_SCORE  (OFF_M2 + (size_t)ROWS*HW)      // 4*HW
#define OFF_G      (OFF_SCORE + (size_t)Bq*HW)     // 4*16*65536
#define OFF_YPRE   (OFF_G + (size_t)Bq*Cq*KSEL)    // 4*16*65536

// ---------------- WMMA (fp32 16x16x4) ----------------
typedef __attribute__((ext_vector_type(2))) float v2f;
typedef __attribute__((ext_vector_type(8))) float v8f;

__device__ __forceinline__ v8f wmma_f32(v2f a, v2f b, v8f c) {
  // D = A(16x4) * B(4x16) + C, fp32 throughout
  return __builtin_amdgcn_wmma_f32_16x16x4_f32(false, a, false, b, (short)0, c, false, false);
}

// ---------------- CDNA5 async global->LDS copy (ASYNCcnt path) ----------------
#if __has_builtin(__builtin_amdgcn_global_load_async_to_lds_b128)
#define HAVE_ASYNC_LDS 1
typedef __attribute__((ext_vector_type(4))) int v4i;
typedef __attribute__((address_space(1))) v4i gv4i;   // global int4
typedef __attribute__((address_space(3))) v4i lv4i;   // LDS int4
__device__ __forceinline__ void asyncCopy16(const float* g, float* l) {
  __builtin_amdgcn_global_load_async_to_lds_b128((gv4i*)g, (lv4i*)l, 0, 0);
}
__device__ __forceinline__ void asyncWait0() {
#if __has_builtin(__builtin_amdgcn_s_wait_asynccnt)
  __builtin_amdgcn_s_wait_asynccnt(0);
#else
  asm volatile("s_wait_asynccnt 0x0" ::: "memory");
#endif
}
#endif

__device__ __forceinline__ unsigned keyOf(float f) {
  unsigned u = __float_as_uint(f);
  return (u & 0x80000000u) ? ~u : (u | 0x80000000u); // order-preserving, larger float -> larger key
}

// ---------------- 0: init / zero scratch ----------------
__global__ void k_init(float* ws) {
  size_t i = (size_t)blockIdx.x * 256 + threadIdx.x;
  if (i < 4096) ws[OFF_ACC1 + i] = 0.f;
  if (i < 128)  ws[OFF_RSUM + i] = 0.f;       // covers RSUM+RSQ
  if (i < 4096) ws[OFF_HIST + i] = 0.f;       // bit-zero == u32 zero
  if (i < 32)   ws[OFF_BNS  + i] = 0.f;
  if (i < 4) {
    unsigned* st = (unsigned*)(ws + OFF_STATE);
    st[i*4+0] = 0u;             // prefix
    st[i*4+1] = (unsigned)KSEL; // kRem
    st[i*4+2] = 0u;             // count(> T) accumulator
    st[i*4+3] = 0u;
  }
}

// ---------------- 1: Linear1 split-K WMMA with LDS staging ----------------
// acc[64][64] += m[64][Kslice] * W1[64][Kslice]^T, one 64x64 tile per block,
// A/B slices staged to LDS once per block (async global->LDS when available)
// so the 128 MiB of m+W1 is read exactly once across the grid.
#define KCHUNK 2048
#define KS     128                      // k-slice staged in LDS per iteration
__global__ void __launch_bounds__(512) k_lin1(const float* __restrict__ m,
                                              const float* __restrict__ w1,
                                              float* __restrict__ acc) {
  __shared__ float lA[64*KS];           // 32 KiB
  __shared__ float lB[64*KS];           // 32 KiB
  const int tid  = threadIdx.x;
  const int wave = tid >> 5;            // 16 waves
  const int lane = tid & 31;
  const int tm = wave >> 2, tn = wave & 3;
  const int hi = lane >> 4, lo = lane & 15;
  const size_t kbase = (size_t)blockIdx.x * KCHUNK;
  v8f c = {};
  for (int s = 0; s < KCHUNK/KS; ++s) {
    const size_t k0 = kbase + (size_t)s*KS;
    // stage A (rows of m) and B (rows of W1): 64 x KS floats each,
    // 2048 16-byte chunks per matrix, 4 per thread
    #pragma unroll
    for (int it = 0; it < (64*(KS/4))/512; ++it) {
      int cidx = it*512 + tid;
      int row  = cidx >> 5;             // KS/4 = 32 chunks per row
      int seg  = cidx & 31;
      const float* gA = m  + (size_t)row*HW + k0 + (size_t)seg*4;
      const float* gB = w1 + (size_t)row*HW + k0 + (size_t)seg*4;
#ifdef HAVE_ASYNC_LDS
      asyncCopy16(gA, &lA[cidx*4]);
      asyncCopy16(gB, &lB[cidx*4]);
#else
      *(float4*)&lA[cidx*4] = *(const float4*)gA;
      *(float4*)&lB[cidx*4] = *(const float4*)gB;
#endif
    }
#ifdef HAVE_ASYNC_LDS
    asyncWait0();
#endif
    __syncthreads();
    const float* aR = &lA[(tm*16 + lo)*KS];
    const float* bR = &lB[(tn*16 + lo)*KS];
    #pragma unroll
    for (int kk = 0; kk < KS; kk += 4) {
      v2f a = *(const v2f*)(aR + kk + 2*hi);
      v2f b = *(const v2f*)(bR + kk + 2*hi);
      c = wmma_f32(a, b, c);
    }
    __syncthreads();
  }
  #pragma unroll
  for (int v = 0; v < 8; ++v)
    atomicAdd(&acc[(size_t)(tm*16 + v + 8*hi)*64 + tn*16 + lo], c[v]);
}

// ---------------- 2: LN1 + ReLU over 64 cols per row -> h[64][64] ----------------
__global__ void k_ln1(const float* __restrict__ acc, const float* __restrict__ b1,
                      const float* __restrict__ g, const float* __restrict__ bb,
                      float* __restrict__ h) {
  int row = threadIdx.x; // 64 threads
  float s = 0.f, s2 = 0.f;
  for (int j = 0; j < 64; ++j) {
    float v = acc[row*64 + j] + b1[j];
    s += v; s2 += v*v;
  }
  float mu = s * (1.f/64.f);
  float var = s2 * (1.f/64.f) - mu*mu;
  float rstd = rsqrtf(var + EPSF);
  for (int j = 0; j < 64; ++j) {
    float v = acc[row*64 + j] + b1[j];
    v = (v - mu) * rstd * g[j] + bb[j];
    h[row*64 + j] = fmaxf(v, 0.f);
  }
}

// ---------------- 3: Linear2 WMMA: m2pre[64][HW] = h[64][64] * W2^T + b2 ----------------
__global__ void k_lin2(const float* __restrict__ h, const float* __restrict__ w2,
                       const float* __restrict__ b2, float* __restrict__ m2) {
  const int wave = threadIdx.x >> 5;   // 8 waves
  const int lane = threadIdx.x & 31;
  const int tm = wave & 3;
  const int ntile = blockIdx.x * 2 + (wave >> 2);
  const int hi = lane >> 4, lo = lane & 15;
  const float* aRow = h  + (size_t)(tm*16 + lo) * 64;
  const float* bRow = w2 + (size_t)(ntile*16 + lo) * 64;  // W2[n][k]
  v8f c = {};
  #pragma unroll
  for (int k0 = 0; k0 < 64; k0 += 4) {
    v2f a = *(const v2f*)(aRow + k0 + 2*hi);
    v2f b = *(const v2f*)(bRow + k0 + 2*hi);
    c = wmma_f32(a, b, c);
  }
  const int n = ntile*16 + lo;
  float bias = b2[n];
  #pragma unroll
  for (int v = 0; v < 8; ++v) {
    int row = tm*16 + v + 8*hi;
    m2[(size_t)row*HW + n] = c[v] + bias;
  }
}

// ---------------- 4: per-row sum / sumsq over HW for LN2 ----------------
__global__ void k_rowred(const float* __restrict__ m2, float* __restrict__ rsum,
                         float* __restrict__ rsq) {
  __shared__ float ls[256], ls2[256];
  int row = blockIdx.y;
  size_t base = (size_t)row*HW + (size_t)blockIdx.x*1024 + (size_t)threadIdx.x*4;
  float4 v = *(const float4*)(m2 + base);
  float s  = v.x + v.y + v.z + v.w;
  float s2 = v.x*v.x + v.y*v.y + v.z*v.z + v.w*v.w;
  ls[threadIdx.x] = s; ls2[threadIdx.x] = s2;
  __syncthreads();
  for (int off = 128; off > 0; off >>= 1) {
    if (threadIdx.x < off) { ls[threadIdx.x] += ls[threadIdx.x+off]; ls2[threadIdx.x] += ls2[threadIdx.x+off]; }
    __syncthreads();
  }
  if (threadIdx.x == 0) { atomicAdd(&rsum[row], ls[0]); atomicAdd(&rsq[row], ls2[0]); }
}

// ---------------- 5: LN2 + ReLU + 1x1 conv -> score[B][HW] ----------------
__global__ void k_ln2score(const float* __restrict__ m2, const float* __restrict__ rsum,
                           const float* __restrict__ rsq, const float* __restrict__ g,
                           const float* __restrict__ bb, const float* __restrict__ cw1,
                           const float* __restrict__ cb1, float* __restrict__ score) {
  int b = blockIdx.y;
  size_t p = (size_t)blockIdx.x*256 + threadIdx.x;
  float gg = g[p], bbb = bb[p];
  float s = cb1[0];
  #pragma unroll
  for (int c = 0; c < Cq; ++c) {
    int row = b*Cq + c;
    float mu  = rsum[row] * (1.f/(float)HW);
    float var = rsq[row]  * (1.f/(float)HW) - mu*mu;
    float rstd = rsqrtf(var + EPSF);
    float v = (m2[(size_t)row*HW + p] - mu) * rstd * gg + bbb;
    v = fmaxf(v, 0.f);
    s += v * cw1[c];
  }
  score[(size_t)b*HW + p] = s;
}

// ---------------- 6: radix-select histogram pass ----------------
__global__ void k_hist(const float* __restrict__ score, const unsigned* __restrict__ state,
                       unsigned* __restrict__ hist, int shift) {
  __shared__ unsigned lh[256];
  lh[threadIdx.x] = 0u;
  __syncthreads();
  int b = blockIdx.y;
  unsigned prefix = state[b*4+0];
  size_t base = (size_t)b*HW + (size_t)blockIdx.x*1024 + (size_t)threadIdx.x*4;
  #pragma unroll
  for (int i = 0; i < 4; ++i) {
    unsigned k = keyOf(score[base + i]);
    if ((((unsigned long long)(k ^ prefix)) >> (shift + 8)) == 0ull)
      atomicAdd(&lh[(k >> shift) & 255u], 1u);
  }
  __syncthreads();
  if (lh[threadIdx.x]) atomicAdd(&hist[b*256 + threadIdx.x], lh[threadIdx.x]);
}

// ---------------- 7: radix-select bin choice ----------------
__global__ void k_select(const unsigned* __restrict__ hist, unsigned* __restrict__ state, int shift) {
  int b = threadIdx.x;
  if (b >= Bq) return;
  unsigned prefix = state[b*4+0], kRem = state[b*4+1], gacc = state[b*4+2];
  unsigned cum = 0, greater = 0; int chosen = 0;
  for (int bin = 255; bin >= 0; --bin) {
    unsigned c = hist[b*256 + bin];
    if (cum + c >= kRem) { chosen = bin; greater = cum; break; }
    cum += c;
  }
  prefix |= ((unsigned)chosen) << shift;
  kRem -= greater; gacc += greater;
  state[b*4+0] = prefix; state[b*4+1] = kRem; state[b*4+2] = gacc;
}

// ---------------- 8: per-block >T / ==T counts ----------------
__global__ void k_blkcount(const float* __restrict__ score, const unsigned* __restrict__ state,
                           unsigned* __restrict__ cgt, unsigned* __restrict__ ceq) {
  __shared__ unsigned sg[256], se[256];
  int b = blockIdx.y;
  unsigned T = state[b*4+0];
  size_t base = (size_t)b*HW + (size_t)blockIdx.x*1024 + (size_t)threadIdx.x*4;
  unsigned g = 0, e = 0;
  #pragma unroll
  for (int i = 0; i < 4; ++i) {
    unsigned k = keyOf(score[base + i]);
    g += (k > T); e += (k == T);
  }
  sg[threadIdx.x] = g; se[threadIdx.x] = e;
  __syncthreads();
  for (int off = 128; off > 0; off >>= 1) {
    if (threadIdx.x < off) { sg[threadIdx.x] += sg[threadIdx.x+off]; se[threadIdx.x] += se[threadIdx.x+off]; }
    __syncthreads();
  }
  if (threadIdx.x == 0) { cgt[b*256 + blockIdx.x] = sg[0]; ceq[b*256 + blockIdx.x] = se[0]; }
}

// ---------------- 9: exclusive scan over 256 blocks per batch ----------------
__global__ void k_blkscan(const unsigned* __restrict__ cgt, const unsigned* __restrict__ ceq,
                          unsigned* __restrict__ pgt, unsigned* __restrict__ peq) {
  int b = threadIdx.x;
  if (b >= Bq) return;
  unsigned rg = 0, re = 0;
  for (int blk = 0; blk < 256; ++blk) {
    pgt[b*256+blk] = rg; peq[b*256+blk] = re;
    rg += cgt[b*256+blk]; re += ceq[b*256+blk];
  }
}

// ---------------- 10: mask write + ranked gather of original m ----------------
__global__ void k_maskgather(const float* __restrict__ input, const float* __restrict__ score,
                             const unsigned* __restrict__ state, const unsigned* __restrict__ pgt,
                             const unsigned* __restrict__ peq, float* __restrict__ mask,
                             float* __restrict__ gout) {
  __shared__ unsigned sg[256], se[256];
  int b = blockIdx.y;
  unsigned T = state[b*4+0];
  unsigned R = state[b*4+1];          // ties to take (ascending index)
  size_t p0 = (size_t)blockIdx.x*1024 + (size_t)threadIdx.x*4;
  unsigned key[4];
  unsigned tg = 0, te = 0;
  #pragma unroll
  for (int i = 0; i < 4; ++i) {
    key[i] = keyOf(score[(size_t)b*HW + p0 + i]);
    tg += (key[i] > T); te += (key[i] == T);
  }
  sg[threadIdx.x] = tg; se[threadIdx.x] = te;
  __syncthreads();
  for (int off = 1; off < 256; off <<= 1) {
    unsigned vg = 0, ve = 0;
    if (threadIdx.x >= (unsigned)off) { vg = sg[threadIdx.x - off]; ve = se[threadIdx.x - off]; }
    __syncthreads();
    sg[threadIdx.x] += vg; se[threadIdx.x] += ve;
    __syncthreads();
  }
  unsigned gBefore = pgt[b*256 + blockIdx.x] + sg[threadIdx.x] - tg; // exclusive
  unsigned eBefore = peq[b*256 + blockIdx.x] + se[threadIdx.x] - te;
  #pragma unroll
  for (int i = 0; i < 4; ++i) {
    bool isG = key[i] > T, isE = key[i] == T;
    bool sel = isG || (isE && eBefore < R);
    size_t p = p0 + i;
    mask[(size_t)b*HW + p] = sel ? 1.f : 0.f;
    if (sel) {
      unsigned rank = gBefore + (eBefore < R ? eBefore : R);
      #pragma unroll
      for (int c = 0; c < Cq; ++c)
        gout[(size_t)(b*Cq + c)*KSEL + rank] = input[(size_t)(b*Cq + c)*HW + p];
    }
    gBefore += isG ? 1u : 0u;
    eBefore += isE ? 1u : 0u;
  }
}

// ---------------- 11: 3x3 conv (16->16) + bias + BN partial stats ----------------
__global__ void k_conv(const float* __restrict__ g, const float* __restrict__ w,
                       const float* __restrict__ cb, float* __restrict__ ypre,
                       float* __restrict__ bns) {
  __shared__ float wl[OUTC*Cq*9];   // 2304
  __shared__ float tile[18*18];
  __shared__ float bs[OUTC], bs2[OUTC];
  int tx = threadIdx.x, ty = threadIdx.y;
  int tid = ty*16 + tx;
  int b = blockIdx.z;
  int ox0 = blockIdx.x*16, oy0 = blockIdx.y*16;
  for (int i = tid; i < OUTC*Cq*9; i += 256) wl[i] = w[i];
  if (tid < OUTC) { bs[tid] = 0.f; bs2[tid] = 0.f; }
  float acc[OUTC];
  #pragma unroll
  for (int oc = 0; oc < OUTC; ++oc) acc[oc] = 0.f;
  for (int ic = 0; ic < Cq; ++ic) {
    __syncthreads();
    for (int i = tid; i < 324; i += 256) {
      int ly = i / 18, lx = i % 18;
      int gy = oy0 + ly - 1, gx = ox0 + lx - 1;
      float v = 0.f;
      if (gy >= 0 && gy < OH && gx >= 0 && gx < OW)
        v = g[((size_t)(b*Cq + ic))*(OH*OW) + (size_t)gy*OW + gx];
      tile[i] = v;
    }
    __syncthreads();
    #pragma unroll
    for (int ky = 0; ky < 3; ++ky)
      #pragma unroll
      for (int kx = 0; kx < 3; ++kx) {
        float v = tile[(ty+ky)*18 + tx+kx];
        #pragma unroll
        for (int oc = 0; oc < OUTC; ++oc)
          acc[oc] += v * wl[(oc*Cq + ic)*9 + ky*3 + kx];
      }
  }
  __syncthreads();
  #pragma unroll
  for (int oc = 0; oc < OUTC; ++oc) {
    float y = acc[oc] + cb[oc];
    ypre[((size_t)(b*OUTC + oc))*(OH*OW) + (size_t)(oy0+ty)*OW + ox0+tx] = y;
    atomicAdd(&bs[oc], y);
    atomicAdd(&bs2[oc], y*y);
  }
  __syncthreads();
  if (tid < OUTC) {
    atomicAdd(&bns[tid], bs[tid]);
    atomicAdd(&bns[OUTC + tid], bs2[tid]);
  }
}

// ---------------- 12: BN finalize + ReLU -> d_out y ----------------
__global__ void k_bn(const float* __restrict__ ypre, const float* __restrict__ bns,
                     const float* __restrict__ bg, const float* __restrict__ bb,
                     float* __restrict__ out) {
  size_t idx = (size_t)blockIdx.x*256 + threadIdx.x; // over 4*16*65536
  int oc = (int)((idx >> 16) & 15);
  const float invN = 1.f / (float)(Bq*OH*OW);
  float mu  = bns[oc] * invN;
  float var = bns[OUTC + oc] * invN - mu*mu;
  float v = (ypre[idx] - mu) * rsqrtf(var + EPSF) * bg[oc] + bb[oc];
  out[idx] = fmaxf(v, 0.f);
}

// ---------------- launch ----------------
extern "C" void kernel_launch(void* const* d_in, const int* in_sizes, int n_in,
                              void* d_out, int out_size, void* d_ws, size_t ws_size,
                              hipStream_t stream) {
  const float* input = (const float*)d_in[0];
  const float* W1    = (const float*)d_in[1];
  const float* b1    = (const float*)d_in[2];
  const float* ln1g  = (const float*)d_in[3];
  const float* ln1b  = (const float*)d_in[4];
  const float* W2    = (const float*)d_in[5];
  const float* b2    = (const float*)d_in[6];
  const float* ln2g  = (const float*)d_in[7];
  const float* ln2b  = (const float*)d_in[8];
  const float* cw1   = (const float*)d_in[9];
  const float* cb1   = (const float*)d_in[10];
  const float* cw3   = (const float*)d_in[11];
  const float* cb3   = (const float*)d_in[12];
  const float* bng   = (const float*)d_in[13];
  const float* bnb   = (const float*)d_in[14];

  float* ws   = (float*)d_ws;
  float* yOut = (float*)d_out;                       // [4,16,256,256]
  float* mOut = yOut + (size_t)Bq*OUTC*OH*OW;        // [4,1,512,512]

  float*    acc1  = ws + OFF_ACC1;
  float*    h     = ws + OFF_H;
  float*    rsum  = ws + OFF_RSUM;
  float*    rsq   = ws + OFF_RSQ;
  unsigned* hist  = (unsigned*)(ws + OFF_HIST);
  unsigned* state = (unsigned*)(ws + OFF_STATE);
  unsigned* cgt   = (unsigned*)(ws + OFF_CGT);
  unsigned* ceq   = (unsigned*)(ws + OFF_CEQ);
  unsigned* pgt   = (unsigned*)(ws + OFF_PGT);
  unsigned* peq   = (unsigned*)(ws + OFF_PEQ);
  float*    bns   = ws + OFF_BNS;
  float*    m2    = ws + OFF_M2;
  float*    score = ws + OFF_SCORE;
  float*    gbuf  = ws + OFF_G;
  float*    ypre  = ws + OFF_YPRE;

  hipLaunchKernelGGL(k_init, dim3(16), dim3(256), 0, stream, ws);
  hipLaunchKernelGGL(k_lin1, dim3(HW/KCHUNK), dim3(512), 0, stream, input, W1, acc1);
  hipLaunchKernelGGL(k_ln1, dim3(1), dim3(64), 0, stream, acc1, b1, ln1g, ln1b, h);
  hipLaunchKernelGGL(k_lin2, dim3(HW/32), dim3(256), 0, stream, h, W2, b2, m2);
  hipLaunchKernelGGL(k_rowred, dim3(HW/1024, ROWS), dim3(256), 0, stream, m2, rsum, rsq);
  hipLaunchKernelGGL(k_ln2score, dim3(HW/256, Bq), dim3(256), 0, stream,
                     m2, rsum, rsq, ln2g, ln2b, cw1, cb1, score);
  for (int pass = 0; pass < 4; ++pass) {
    int shift = 24 - 8*pass;
    hipLaunchKernelGGL(k_hist, dim3(HW/1024, Bq), dim3(256), 0, stream,
                       score, state, hist + pass*Bq*256, shift);
    hipLaunchKernelGGL(k_select, dim3(1), dim3(32), 0, stream,
                       hist + pass*Bq*256, state, shift);
  }
  hipLaunchKernelGGL(k_blkcount, dim3(HW/1024, Bq), dim3(256), 0, stream, score, state, cgt, ceq);
  hipLaunchKernelGGL(k_blkscan, dim3(1), dim3(32), 0, stream, cgt, ceq, pgt, peq);
  hipLaunchKernelGGL(k_maskgather, dim3(HW/1024, Bq), dim3(256), 0, stream,
                     input, score, state, pgt, peq, mOut, gbuf);
  hipLaunchKernelGGL(k_conv, dim3(OW/16, OH/16, Bq), dim3(16,16), 0, stream,
                     gbuf, cw3, cb3, ypre, bns);
  hipLaunchKernelGGL(k_bn, dim3((Bq*OUTC*OH*OW)/256), dim3(256), 0, stream,
                     ypre, bns, bng, bnb, yOut);
}